// big_net_62371515073166
// MI455X (gfx1250) — compile-verified
//
#include <hip/hip_runtime.h>
#include <math.h>

typedef float v2f __attribute__((ext_vector_type(2)));
typedef float v8f __attribute__((ext_vector_type(8)));

#define IMG  64
#define NPIX 4096     // 64*64
#define CH   16
#define NB   8

// ---------------------------------------------------------------------------
// 5x5 'same' conv, NCHW, zero padding. grid = (16, Cout=16, B=8), block = 256.
// Tiny working set (<= 2 MB), fully L2 resident; perf-irrelevant.
// ---------------------------------------------------------------------------
__global__ void __launch_bounds__(256)
conv5x5_kernel(const float* __restrict__ in, const float* __restrict__ w,
               const float* __restrict__ bias, float* __restrict__ out, int Cin) {
  const int b  = blockIdx.z;
  const int co = blockIdx.y;
  const int p  = blockIdx.x * 256 + threadIdx.x;   // pixel 0..4095
  const int y  = p >> 6, x = p & 63;
  float acc = bias[co];
  for (int ci = 0; ci < Cin; ++ci) {
    const float* ip = in + (size_t)(b * Cin + ci) * NPIX;
    const float* wp = w  + (size_t)(co * Cin + ci) * 25;
#pragma unroll
    for (int ky = 0; ky < 5; ++ky) {
      const int iy = y + ky - 2;
      if (iy < 0 || iy > 63) continue;
#pragma unroll
      for (int kx = 0; kx < 5; ++kx) {
        const int ix = x + kx - 2;
        if (ix < 0 || ix > 63) continue;
        acc = fmaf(ip[iy * IMG + ix], wp[ky * 5 + kx], acc);
      }
    }
  }
  out[(size_t)(b * CH + co) * NPIX + p] = acc;
}

// ---------------------------------------------------------------------------
// Per (b, pixel): mean/std over 16 channels (unbiased), emit
//   z[b,c,i] = (x3[b,c,i] - mean) / (std * sqrt(C))
// so that coupling = z^T z * mask directly.
// ---------------------------------------------------------------------------
__global__ void __launch_bounds__(256)
center_scale_kernel(const float* __restrict__ x3, float* __restrict__ z) {
  const int gid = blockIdx.x * 256 + threadIdx.x;  // 0 .. 8*4096-1
  const int b = gid >> 12;
  const int i = gid & (NPIX - 1);
  float v[CH];
  float s = 0.f, sq = 0.f;
#pragma unroll
  for (int c = 0; c < CH; ++c) {
    const float t = x3[(size_t)(b * CH + c) * NPIX + i];
    v[c] = t; s += t; sq += t * t;
  }
  const float mean = s * (1.0f / CH);
  float var = (sq - (float)CH * mean * mean) * (1.0f / (CH - 1));
  var = fmaxf(var, 0.0f);
  const float inv = 1.0f / (sqrtf(var) * 4.0f);   // 1/(std*sqrt(16))
#pragma unroll
  for (int c = 0; c < CH; ++c)
    z[(size_t)(b * CH + c) * NPIX + i] = (v[c] - mean) * inv;
}

// ---------------------------------------------------------------------------
// coupling[b] = (Z^T Z) .* mask, Z: (16 x 4096) fp32, via V_WMMA_F32_16X16X4_F32.
// grid = (4096/128, 4096/128, 8), block = 256 (8 waves).
// Wave w computes rows [ (by*8+w)*16 , +16 ) x cols [ bx*128, +128 ) as 8 tiles.
// K=16 -> 4 chained WMMAs per tile. EXEC is all-ones (no divergence, exact grid).
// A (16x4 f32): lanes 0-15 = M, v.x/v.y = K,K+1; lane-half selects K+0 / K+2.
// B (4x16 f32): same striping with N in lanes. D: row = r + 8*(lane>=16).
// Output stream (512 MB) stored non-temporally to keep the mask in L2.
// ---------------------------------------------------------------------------
__global__ void __launch_bounds__(256)
coupling_wmma_kernel(const float* __restrict__ z, const float* __restrict__ mask,
                     float* __restrict__ out) {
  const int b    = blockIdx.z;
  const int wave = threadIdx.x >> 5;
  const int lane = threadIdx.x & 31;
  const int m    = lane & 15;
  const int kh   = (lane >> 4) << 1;          // 0 or 2
  const int i0   = (blockIdx.y * 8 + wave) * 16;
  const int j0   = blockIdx.x * 128;
  const int rowo = (lane >> 4) << 3;          // 0 or 8

  const float* zb = z + (size_t)b * CH * NPIX;

  v2f a[4];
#pragma unroll
  for (int kc = 0; kc < 4; ++kc) {
    const int k = kc * 4 + kh;
    a[kc].x = zb[(size_t)(k + 0) * NPIX + i0 + m];
    a[kc].y = zb[(size_t)(k + 1) * NPIX + i0 + m];
  }

#pragma unroll
  for (int jt = 0; jt < 8; ++jt) {
    const int j = j0 + jt * 16;
    v2f bv[4];
#pragma unroll
    for (int kc = 0; kc < 4; ++kc) {
      const int k = kc * 4 + kh;
      bv[kc].x = zb[(size_t)(k + 0) * NPIX + j + m];
      bv[kc].y = zb[(size_t)(k + 1) * NPIX + j + m];
    }
    v8f d = {};
#pragma unroll
    for (int kc = 0; kc < 4; ++kc)
      d = __builtin_amdgcn_wmma_f32_16x16x4_f32(
          /*neg_a=*/false, a[kc], /*neg_b=*/false, bv[kc],
          /*c_mod=*/(short)0, d, /*reuse_a=*/false, /*reuse_b=*/false);

    const int col = j + m;
#pragma unroll
    for (int r = 0; r < 8; ++r) {
      const int row = i0 + r + rowo;
      const float val = d[r] * mask[(size_t)row * NPIX + col];
      __builtin_nontemporal_store(val, &out[((size_t)b * NPIX + row) * NPIX + col]);
    }
  }
}

// ---------------------------------------------------------------------------
extern "C" void kernel_launch(void* const* d_in, const int* in_sizes, int n_in,
                              void* d_out, int out_size, void* d_ws, size_t ws_size,
                              hipStream_t stream) {
  const float* x    = (const float*)d_in[0];
  const float* w0   = (const float*)d_in[1];
  const float* b0   = (const float*)d_in[2];
  const float* w1   = (const float*)d_in[3];
  const float* b1   = (const float*)d_in[4];
  const float* w2   = (const float*)d_in[5];
  const float* b2   = (const float*)d_in[6];
  const float* mask = (const float*)d_in[7];
  float*       out  = (float*)d_out;

  float* bufA = (float*)d_ws;                       // 8*16*4096 floats (2 MB)
  float* bufB = bufA + (size_t)NB * CH * NPIX;      // 8*16*4096 floats (2 MB)

  const dim3 cblk(256), cgrd(NPIX / 256, CH, NB);
  conv5x5_kernel<<<cgrd, cblk, 0, stream>>>(x,    w0, b0, bufA, 1);   // x -> x1
  conv5x5_kernel<<<cgrd, cblk, 0, stream>>>(bufA, w1, b1, bufB, CH);  // x1 -> x2
  conv5x5_kernel<<<cgrd, cblk, 0, stream>>>(bufB, w2, b2, bufA, CH);  // x2 -> x3

  center_scale_kernel<<<dim3(NB * NPIX / 256), 256, 0, stream>>>(bufA, bufB);

  coupling_wmma_kernel<<<dim3(NPIX / 128, NPIX / 128, NB), 256, 0, stream>>>(
      bufB, mask, out);
}